// FCC_Network_76484777607417
// MI455X (gfx1250) — compile-verified
//
#include <hip/hip_runtime.h>
#include <hip/hip_bf16.h>

typedef __attribute__((ext_vector_type(2))) float v2f;
typedef __attribute__((ext_vector_type(8))) float v8f;

// Problem constants (fixed at module scope in the reference)
#define TRUN 2
#define BSZ  16
#define NOPT 4
#define HDIM 1024
#define SLEN 1536

// ---------------------------------------------------------------------------
// Kernel 1: gather + average -> feats [64, 1024]
// feats[b*4+o, h] = 0.25 * sum_t ( e[t,b,0,h]+e[t,b,511,h]          (p)
//                                + e[t,b,512,h]+e[t,b,639,h]        (q)
//                                + e[t,b,oh[o],h]+e[t,b,ot[o],h] )  (opt)
// (0.25 = 0.5 * 1/TRUN with TRUN=2, bi_direction=1)
// ---------------------------------------------------------------------------
__global__ __launch_bounds__(256) void feats_kernel(const float* __restrict__ emb,
                                                    float* __restrict__ feats) {
    const int r = blockIdx.x;           // 0..63  (= b*NOPT + o)
    const int b = r >> 2;
    const int o = r & 3;
    const int oHead[4] = {640, 768, 896, 1024};
    const int oTail[4] = {767, 895, 1023, 1151};
    const int h0 = oHead[o], t0 = oTail[o];

    for (int h = threadIdx.x; h < HDIM; h += 256) {
        float acc = 0.0f;
#pragma unroll
        for (int t = 0; t < TRUN; ++t) {
            const float* e = emb + ((size_t)(t * BSZ + b)) * SLEN * HDIM + h;
            acc += e[(size_t)0    * HDIM] + e[(size_t)511 * HDIM]
                 + e[(size_t)512  * HDIM] + e[(size_t)639 * HDIM]
                 + e[(size_t)h0   * HDIM] + e[(size_t)t0  * HDIM];
        }
        feats[r * HDIM + h] = 0.25f * acc;
    }
}

// ---------------------------------------------------------------------------
// GEMM helpers: fragment stage = 4 k-steps (K16) of V_WMMA_F32_16X16X4_F32.
// Fragment layouts per CDNA5 ISA 7.12.2:
//   A (16x4): lane<16 holds {K=0,1} of row M=lane; lane>=16 holds {K=2,3}.
//   B (4x16): lane<16 holds {K=0,1} of col N=lane; lane>=16 holds {K=2,3}.
//   C/D     : VGPR r, lanes0-15 -> M=r, lanes16-31 -> M=8+r; N=lane&15.
// ---------------------------------------------------------------------------
__device__ __forceinline__ void load_stage(const float* __restrict__ Arow,
                                           const float* __restrict__ Wcol,
                                           int kbase, int half,
                                           v2f* __restrict__ a, v2f* __restrict__ b) {
#pragma unroll
    for (int u = 0; u < 4; ++u) {
        const int k = kbase + 4 * u;
        a[u] = *(const v2f*)(Arow + k);             // A[m][k+2h], A[m][k+2h+1]
        const int krow = k + 2 * half;
        b[u].x = Wcol[(size_t)krow * HDIM];         // W[k+2h  ][n]
        b[u].y = Wcol[(size_t)(krow + 1) * HDIM];   // W[k+2h+1][n]
    }
}

__device__ __forceinline__ void wmma_stage(const v2f* __restrict__ a,
                                           const v2f* __restrict__ b, v8f& acc) {
#pragma unroll
    for (int u = 0; u < 4; ++u)
        acc = __builtin_amdgcn_wmma_f32_16x16x4_f32(
                  /*neg_a=*/false, a[u], /*neg_b=*/false, b[u],
                  /*c_mod=*/(short)0, acc, /*reuse_a=*/false, /*reuse_b=*/false);
}

// out = relu(A @ W + bias), A:[64,K=1024], W:[1024,1024], f32.
// One wave (EXEC all-1s, WMMA requirement) per 16x16 tile; software-pipelined
// double-buffered fragments so loads of stage k+16 overlap WMMAs of stage k.
__global__ __launch_bounds__(32) void gemm_relu_kernel(const float* __restrict__ A,
                                                       const float* __restrict__ W,
                                                       const float* __restrict__ bias,
                                                       float* __restrict__ out) {
    const int lane = threadIdx.x;       // 0..31
    const int n0   = blockIdx.x * 16;   // 64 N-tiles
    const int m0   = blockIdx.y * 16;   // 4  M-tiles
    const int half = lane >> 4;         // 0 | 1 -> selects K pair
    const int l    = lane & 15;

    const int K = HDIM, N = HDIM;
    const float* __restrict__ Arow = A + (size_t)(m0 + l) * K + 2 * half;
    const float* __restrict__ Wcol = W + n0 + l;

    // one unguarded stream hint; W1/W2 are L2-resident after first touch
    __builtin_prefetch(Wcol, 0, 1);

    v8f acc = {};
    v2f a0[4], b0[4], a1[4], b1[4];

    load_stage(Arow, Wcol, 0, half, a0, b0);
    int kbase = 0;
    for (; kbase < K - 32; kbase += 32) {
        load_stage(Arow, Wcol, kbase + 16, half, a1, b1);
        wmma_stage(a0, b0, acc);
        load_stage(Arow, Wcol, kbase + 32, half, a0, b0);
        wmma_stage(a1, b1, acc);
    }
    // epilogue: kbase == K-32
    load_stage(Arow, Wcol, kbase + 16, half, a1, b1);
    wmma_stage(a0, b0, acc);
    wmma_stage(a1, b1, acc);

    const float bv = bias[n0 + l];
#pragma unroll
    for (int r = 0; r < 8; ++r) {
        const int row = m0 + r + 8 * half;
        out[(size_t)row * N + (n0 + l)] = fmaxf(acc[r] + bv, 0.0f);
    }
}

// ---------------------------------------------------------------------------
// Kernel 4: result = h2 @ W3  ([64,1024] @ [1024]) ; loss = NLL(log_softmax)
// d_out: result[64] then loss[1]
// ---------------------------------------------------------------------------
__global__ __launch_bounds__(256) void head_kernel(const float* __restrict__ h2,
                                                   const float* __restrict__ W3,
                                                   const int* __restrict__ y,
                                                   float* __restrict__ d_out) {
    __shared__ float partial[256];
    __shared__ float res[64];
    const int tid  = threadIdx.x;
    const int row  = tid >> 2;          // 64 rows, 4 threads per row
    const int part = tid & 3;

    const float* hrow = h2 + (size_t)row * HDIM;
    float s = 0.0f;
    const int k0 = part * (HDIM / 4);
    for (int k = k0; k < k0 + HDIM / 4; ++k) s += hrow[k] * W3[k];
    partial[tid] = s;
    __syncthreads();

    if (part == 0) {
        float v = partial[tid] + partial[tid + 1] + partial[tid + 2] + partial[tid + 3];
        res[row]   = v;
        d_out[row] = v;                 // result [B, NOPT] row-major
    }
    __syncthreads();

    if (tid == 0) {
        float loss = 0.0f;
        for (int b = 0; b < BSZ; ++b) {
            const float* rr = res + b * NOPT;
            float m = fmaxf(fmaxf(rr[0], rr[1]), fmaxf(rr[2], rr[3]));
            float se = 0.0f;
            for (int o = 0; o < NOPT; ++o) se += expf(rr[o] - m);
            const float lse = m + logf(se);
            int lbl = y[b];
            if (lbl < 0) lbl = 0;
            if (lbl >= NOPT) lbl = NOPT - 1;
            loss += (lse - rr[lbl]);
        }
        d_out[64] = loss / (float)BSZ;
    }
}

// ---------------------------------------------------------------------------
extern "C" void kernel_launch(void* const* d_in, const int* in_sizes, int n_in,
                              void* d_out, int out_size, void* d_ws, size_t ws_size,
                              hipStream_t stream) {
    const float* emb = (const float*)d_in[0];   // [TRUN*B, S, H] f32
    const float* W1  = (const float*)d_in[1];   // [H, H]
    const float* b1  = (const float*)d_in[2];   // [H]
    const float* W2  = (const float*)d_in[3];   // [H, H]
    const float* b2  = (const float*)d_in[4];   // [H]
    const float* W3  = (const float*)d_in[5];   // [H, 1]
    const int*   y   = (const int*)d_in[6];     // [B] int32
    float*       out = (float*)d_out;           // result[64] + loss[1]

    float* feats = (float*)d_ws;                        // 64*1024 f32
    float* h1    = feats + 64 * HDIM;                   // 64*1024 f32
    float* h2    = h1    + 64 * HDIM;                   // 64*1024 f32

    feats_kernel<<<64, 256, 0, stream>>>(emb, feats);

    dim3 ggrid(HDIM / 16, 64 / 16);                     // 64 N-tiles x 4 M-tiles
    gemm_relu_kernel<<<ggrid, 32, 0, stream>>>(feats, W1, b1, h1);
    gemm_relu_kernel<<<ggrid, 32, 0, stream>>>(h1,    W2, b2, h2);

    head_kernel<<<1, 256, 0, stream>>>(h2, W3, y, out);
}